// STPPTest_75179107549592
// MI455X (gfx1250) — compile-verified
//
#include <hip/hip_runtime.h>
#include <hip/hip_bf16.h>

// ---------------------------------------------------------------------------
// Problem constants (from reference)
// ---------------------------------------------------------------------------
#define NUM_CLASSES 200
#define ACT_LEN     (NUM_CLASSES + 1)        // 201
#define COMP_LEN    (NUM_CLASSES)            // 200
#define REG_LEN     (NUM_CLASSES * 2)        // 400
#define MULT        5
#define FEAT_DIM    (ACT_LEN + MULT * (COMP_LEN + REG_LEN))   // 3201
#define T_LEN       16384
#define N_PROP      2048

#define CHUNK       16                        // rows per scan chunk (== WMMA tile M)
#define NCH         (T_LEN / CHUNK)           // 1024
#define FEAT_PAD    3328                      // 13 * 256, multiple of 16
#define COL_TILES   (FEAT_PAD / 16)           // 208

typedef float v2f __attribute__((ext_vector_type(2)));
typedef float v8f __attribute__((ext_vector_type(8)));

// ---------------------------------------------------------------------------
// Kernel A: per-chunk column sums.  chunk[c][col] = sum_{j<16} x[c*16+j][col]
// ---------------------------------------------------------------------------
__global__ void stpp_chunk_sums(const float* __restrict__ x,
                                float* __restrict__ chunk)
{
    const int col = blockIdx.y * blockDim.x + threadIdx.x;   // padded col
    const int c   = blockIdx.x;                              // chunk id
    if (col >= FEAT_PAD) return;
    float s = 0.0f;
    if (col < FEAT_DIM) {
        const float* p = x + (size_t)c * CHUNK * FEAT_DIM + col;
        // speculative prefetch of the next chunk's first row (gfx1250 global_prefetch)
        __builtin_prefetch(p + (size_t)CHUNK * FEAT_DIM, 0, 0);
#pragma unroll
        for (int j = 0; j < CHUNK; ++j)
            s += p[(size_t)j * FEAT_DIM];
    }
    chunk[(size_t)c * FEAT_PAD + col] = s;
}

// ---------------------------------------------------------------------------
// Kernel B: in-place exclusive scan of chunk sums along the chunk axis.
// One thread per column (3328 threads); 1024 sequential adds each — tiny.
// ---------------------------------------------------------------------------
__global__ void stpp_scan_chunks(float* __restrict__ chunk)
{
    const int col = blockIdx.x * blockDim.x + threadIdx.x;
    if (col >= FEAT_PAD) return;
    float run = 0.0f;
    for (int c = 0; c < NCH; ++c) {
        const size_t idx = (size_t)c * FEAT_PAD + col;
        const float t = chunk[idx];
        chunk[idx] = run;           // exclusive prefix
        run += t;
    }
}

// ---------------------------------------------------------------------------
// Kernel C0: cs row 0 = 0
// ---------------------------------------------------------------------------
__global__ void stpp_zero_row0(float* __restrict__ cs)
{
    const int col = blockIdx.x * blockDim.x + threadIdx.x;
    if (col < FEAT_PAD) cs[col] = 0.0f;
}

// ---------------------------------------------------------------------------
// Kernel C: finalize cumsum with WMMA.
//   For a 16x16 tile (16 timesteps x 16 cols):  D = L * X  with L lower-tri
//   ones => D[m][n] = sum_{k<=m} x[k][n]  (inclusive scan), done as four
//   chained V_WMMA_F32_16X16X4_F32 accumulations over K.
//
// VGPR layouts (CDNA5 ISA 7.12.2, wave32):
//   A 16x4 f32 : lane -> M = lane&15 ; VGPR v, half h=lane>>4 -> K = v + 2h
//   B  4x16 f32: lane -> N = lane&15 ; VGPR v, half h        -> K = v + 2h
//   C/D 16x16  : lane -> N = lane&15 ; VGPR v, half h        -> M = v + 8h
// One wave handles one (16-row group, 16-col tile). EXEC stays all-ones:
// column tail is handled by address clamp + zero mask, not divergence.
// ---------------------------------------------------------------------------
__global__ void stpp_scan16_wmma(const float* __restrict__ x,
                                 const float* __restrict__ chunk,   // exclusive prefixes
                                 float* __restrict__ cs)            // [T+1][FEAT_PAD]
{
    const int lane = threadIdx.x & 31;
    const int wave = threadIdx.x >> 5;
    const int half = lane >> 4;          // 0 | 1
    const int m    = lane & 15;          // M for A, N for B/C/D
    const int ct   = blockIdx.x;                 // column tile
    const int g    = blockIdx.y * 8 + wave;      // 16-row group, 0..NCH-1
    const int colp = ct * 16 + m;                // padded column
    const int colc = (colp < FEAT_DIM) ? colp : (FEAT_DIM - 1);
    const float cmask = (colp < FEAT_DIM) ? 1.0f : 0.0f;
    const int rowbase = g * CHUNK;

    v8f c = {0.f, 0.f, 0.f, 0.f, 0.f, 0.f, 0.f, 0.f};
#pragma unroll
    for (int s = 0; s < 4; ++s) {
        const int k0 = s * 4 + 2 * half;         // global K of VGPR0 element
        v2f b;
        b.x = cmask * x[(size_t)(rowbase + k0 + 0) * FEAT_DIM + colc];
        b.y = cmask * x[(size_t)(rowbase + k0 + 1) * FEAT_DIM + colc];
        v2f a;
        a.x = (k0 + 0 <= m) ? 1.0f : 0.0f;       // lower-triangular ones
        a.y = (k0 + 1 <= m) ? 1.0f : 0.0f;
        c = __builtin_amdgcn_wmma_f32_16x16x4_f32(
                /*neg_a=*/false, a, /*neg_b=*/false, b,
                /*c_mod=*/(short)0, c, /*reuse_a=*/false, /*reuse_b=*/false);
    }

    const float pre = chunk[(size_t)g * FEAT_PAD + colp];   // per-column broadcast
#pragma unroll
    for (int v = 0; v < 8; ++v) {
        const int row = rowbase + v + 8 * half;             // D's M for this VGPR
        cs[(size_t)(row + 1) * FEAT_PAD + colp] = c[v] + pre;
    }
}

// ---------------------------------------------------------------------------
// Kernel D: per-proposal gather.  One block per proposal, threads stride cols.
// Segment parameters follow the reference's f32 arithmetic exactly.
// ---------------------------------------------------------------------------
__device__ __forceinline__ void stpp_seg(int tl_i, int tn_i, float scale,
                                         int n_part, int i,
                                         int& lc, int& rc, float& wt)
{
    const float tl = (float)tl_i;
    const float tr = fmaxf((float)tl_i + 1.0f, (float)tn_i);
    const bool stage_valid = (tr > 0.0f) && (tl < (float)T_LEN);
    const float step = (tr - tl) / (float)n_part;
    const int l = (int)floorf(tl + (float)i * step);
    const int r = (int)floorf(tl + (float)(i + 1) * step);
    const bool valid = stage_valid && (r - l >= 1);
    lc = min(max(l, 0), T_LEN);
    rc = min(max(r, 0), T_LEN);
    const float denom = fmaxf((float)(rc - lc), 1.0f);
    wt = valid ? (scale / denom) : 0.0f;
}

__global__ void stpp_gather(const float* __restrict__ cs,
                            const int* __restrict__ ticks,   // [N,4]
                            const float* __restrict__ sf,    // [N,2]
                            float* __restrict__ out_act,
                            float* __restrict__ out_comp,
                            float* __restrict__ out_reg)
{
    const int n   = blockIdx.x;
    const int tid = threadIdx.x;
    const int t0 = ticks[n * 4 + 0];
    const int t1 = ticks[n * 4 + 1];
    const int t2 = ticks[n * 4 + 2];
    const int t3 = ticks[n * 4 + 3];
    const float s0 = sf[n * 2 + 0];
    const float s1 = sf[n * 2 + 1];

    // ---- act: (cs[r]-cs[l]) / (r-l) over first 201 columns
    {
        const int la = t1;
        const int ra = max(t1 + 1, t2);
        const float wa = 1.0f / (float)(ra - la);
        const float* __restrict__ pr = cs + (size_t)ra * FEAT_PAD;
        const float* __restrict__ pl = cs + (size_t)la * FEAT_PAD;
        for (int c = tid; c < ACT_LEN; c += blockDim.x)
            out_act[(size_t)n * ACT_LEN + c] = wa * (pr[c] - pl[c]);
    }

    // ---- pyramid segments (5 offsets) shared by comp and reg
    int lo[5], ro[5];
    float w[5];
    stpp_seg(t0, t1, s0,   1, 0, lo[0], ro[0], w[0]);   // stage 0
    stpp_seg(t1, t2, 1.0f, 1, 0, lo[1], ro[1], w[1]);   // stage 1, part 1
    stpp_seg(t1, t2, 1.0f, 2, 0, lo[2], ro[2], w[2]);   // stage 1, part 2/0
    stpp_seg(t1, t2, 1.0f, 2, 1, lo[3], ro[3], w[3]);   // stage 1, part 2/1
    stpp_seg(t2, t3, s1,   1, 0, lo[4], ro[4], w[4]);   // stage 2

    // ---- comp: 5 blocks of 200 columns starting at ACT_LEN
    for (int c = tid; c < COMP_LEN; c += blockDim.x) {
        float acc = 0.0f;
#pragma unroll
        for (int o = 0; o < 5; ++o) {
            const size_t cb = (size_t)(ACT_LEN + o * COMP_LEN) + c;
            acc += w[o] * (cs[(size_t)ro[o] * FEAT_PAD + cb]
                         - cs[(size_t)lo[o] * FEAT_PAD + cb]);
        }
        out_comp[(size_t)n * COMP_LEN + c] = acc;
    }

    // ---- reg: 5 blocks of 400 columns starting at ACT_LEN + 5*COMP_LEN
    for (int c = tid; c < REG_LEN; c += blockDim.x) {
        float acc = 0.0f;
#pragma unroll
        for (int o = 0; o < 5; ++o) {
            const size_t cb = (size_t)(ACT_LEN + MULT * COMP_LEN + o * REG_LEN) + c;
            acc += w[o] * (cs[(size_t)ro[o] * FEAT_PAD + cb]
                         - cs[(size_t)lo[o] * FEAT_PAD + cb]);
        }
        out_reg[(size_t)n * REG_LEN + c] = acc;
    }
}

// ---------------------------------------------------------------------------
// Host launcher
// ---------------------------------------------------------------------------
extern "C" void kernel_launch(void* const* d_in, const int* in_sizes, int n_in,
                              void* d_out, int out_size, void* d_ws, size_t ws_size,
                              hipStream_t stream)
{
    const float* x     = (const float*)d_in[0];   // [T, FEAT_DIM] f32
    const int*   ticks = (const int*)d_in[1];     // [N, 4] i32
    const float* sf    = (const float*)d_in[2];   // [N, 2] f32

    // workspace layout
    float* cs    = (float*)d_ws;                                  // (T+1) x FEAT_PAD
    float* chunk = cs + (size_t)(T_LEN + 1) * FEAT_PAD;           // NCH x FEAT_PAD

    float* out_act  = (float*)d_out;                              // [N, 201]
    float* out_comp = out_act + (size_t)N_PROP * ACT_LEN;         // [N, 200]
    float* out_reg  = out_comp + (size_t)N_PROP * COMP_LEN;       // [N, 400]

    const int TB = 256;
    // A: chunk sums
    stpp_chunk_sums<<<dim3(NCH, FEAT_PAD / TB), TB, 0, stream>>>(x, chunk);
    // B: exclusive scan of chunk sums
    stpp_scan_chunks<<<dim3(FEAT_PAD / TB), TB, 0, stream>>>(chunk);
    // C0: zero row of cs
    stpp_zero_row0<<<dim3(FEAT_PAD / TB), TB, 0, stream>>>(cs);
    // C: WMMA-based intra-chunk inclusive scan + prefix add
    stpp_scan16_wmma<<<dim3(COL_TILES, NCH / 8), TB, 0, stream>>>(x, chunk, cs);
    // D: per-proposal gather
    stpp_gather<<<dim3(N_PROP), TB, 0, stream>>>(cs, ticks, sf,
                                                 out_act, out_comp, out_reg);
}